// yololoss_59614146068563
// MI455X (gfx1250) — compile-verified
//
#include <hip/hip_runtime.h>
#include <math.h>

typedef float v2f __attribute__((ext_vector_type(2)));
typedef float v8f __attribute__((ext_vector_type(8)));

#define YOLO_EPS 1e-6f

// ---------------------------------------------------------------------------
// Exact FP32 sum of all 32 lanes using V_WMMA_F32_16X16X4_F32.
// A (16x4, 2 VGPRs): lane L -> row M=L%16; lanes 0-15 hold K=0/1, lanes 16-31
// hold K=2/3. Put partial p in a[0] (K=0 or K=2), 0 in a[1]. B = all ones.
// D[m][n] = sum_k A[m][k] = p(m) + p(m+16), independent of n.
// C/D layout: VGPR j, lanes 0-15 -> M=j, lanes 16-31 -> M=j+8. So summing the
// 8 result VGPRs in a lane gives sum over rows 0-7 (lanes<16) or 8-15
// (lanes>=16); one shfl_xor(16) add yields the full 32-lane sum in every lane.
// Requires EXEC all-1s (callers guarantee convergence at the call site).
// ---------------------------------------------------------------------------
__device__ __forceinline__ float wave_sum32(float p) {
#if defined(__has_builtin) && __has_builtin(__builtin_amdgcn_wmma_f32_16x16x4_f32)
    v2f a; a[0] = p;    a[1] = 0.0f;
    v2f b; b[0] = 1.0f; b[1] = 1.0f;
    v8f c = {};
    c = __builtin_amdgcn_wmma_f32_16x16x4_f32(false, a, false, b,
                                              (short)0, c, false, false);
    float s = ((c[0] + c[1]) + (c[2] + c[3])) + ((c[4] + c[5]) + (c[6] + c[7]));
    s += __shfl_xor(s, 16, 32);
    return s;
#else
    for (int off = 16; off > 0; off >>= 1) p += __shfl_xor(p, off, 32);
    return p;
#endif
}

__device__ __forceinline__ float clamp01f(float x) {
    return fminf(fmaxf(x, 0.0f), 1.0f);
}
__device__ __forceinline__ float clampepsf(float x) {
    return fminf(fmaxf(x, YOLO_EPS), 1.0f);
}

// Faithful translation of the reference _iou (boxes as left, top, w, h).
__device__ __forceinline__ float iou_ref(float l1l, float l1t, float w1, float h1,
                                         float l2l, float l2t, float w2, float h2) {
    float l1r = l1l + w1, l1b = l1t + h1;
    float l2r = l2l + w2, l2b = l2t + h2;
    float a1 = w1 * h1, a2 = w2 * h2;
    bool cont_by2 = (l2l <= l1l) && (l2t <= l1t) && (l2r >= l1r) && (l2b >= l1b);
    bool cont_by1 = (l1l <= l2l) && (l1t <= l2t) && (l1r >= l2r) && (l1b >= l2b);
    bool sep = (l1r <= l2l) || (l2r <= l1l) || (l1b <= l2t) || (l2b <= l1t);
    float iw = fminf(l1r, l2r) - fmaxf(l1l, l2l);
    float ih = fminf(l1b, l2b) - fmaxf(l1t, l2t);
    float inter = iw * ih;
    float gen = inter / (a1 + a2 - inter);
    float r = sep ? 0.0f : gen;
    r = cont_by1 ? (a2 / a1) : r;
    r = cont_by2 ? (a1 / a2) : r;
    return r;
}

// ---------------------------------------------------------------------------
// Pass 1: one thread per (n, i, j) cell. Streams 80B pred + 24B target per
// thread via B128/B64 global loads, computes the per-cell loss contribution,
// reduces per-wave with WMMA, per-block through LDS, writes block partials.
// ---------------------------------------------------------------------------
__global__ void __launch_bounds__(256)
yolo_cell_kernel(const float* __restrict__ pred,
                 const float* __restrict__ targ,
                 float* __restrict__ blk_out,
                 long long ncells) {
    long long cell = (long long)blockIdx.x * 256 + threadIdx.x;
    bool valid = cell < ncells;
    long long c = valid ? cell : (ncells - 1);   // clamp: keep EXEC full

    const float4* pp = reinterpret_cast<const float4*>(pred + c * 20);
    float4 q0 = pp[0];
    float4 q1 = pp[1];
    float4 q2 = pp[2];
    float4 q3 = pp[3];
    float4 q4 = pp[4];
    const float2* tp = reinterpret_cast<const float2*>(targ + c * 6);
    float2 t01 = tp[0];
    float2 t23 = tp[1];
    float2 t45 = tp[2];

    float p0 = q0.x, p1 = q0.y, p2 = q0.z, p3 = q0.w;
    float p4 = q1.x, p5 = q1.y, p6 = q1.z, p7 = q1.w;
    float p8 = q2.x, p9 = q2.y;
    float L[10] = {q2.z, q2.w, q3.x, q3.y, q3.z, q3.w, q4.x, q4.y, q4.z, q4.w};
    float t0 = t01.x, t1 = t01.y, t2 = t23.x, t3 = t23.y;
    float conf_t = t45.x, cls_t = t45.y;

    float coo = (conf_t > 0.0f) ? 1.0f : 0.0f;
    float noo = (conf_t == 0.0f) ? 1.0f : 0.0f;

    // no-object loss
    float d4 = p4 - conf_t, d9 = p9 - conf_t;
    float noobj = noo * (d4 * d4 + d9 * d9);

    // responsible-box selection by IoU
    float iou0 = iou_ref(p0, p1, p2, p3, t0, t1, t2, t3);
    float iou1 = iou_ref(p5, p6, p7, p8, t0, t1, t2, t3);
    bool pick0 = iou0 >= iou1;
    float max_iou = pick0 ? iou0 : iou1;
    float rx = pick0 ? p0 : p5;
    float ry = pick0 ? p1 : p6;
    float rw = pick0 ? p2 : p7;
    float rh = pick0 ? p3 : p8;
    float rc = pick0 ? p4 : p9;
    float nc = pick0 ? p9 : p4;

    float cd = rc - max_iou;
    float contain = coo * cd * cd;

    // localization loss
    float ex = clamp01f(rx) - clamp01f(t0);
    float ey = clamp01f(ry) - clamp01f(t1);
    float ew = sqrtf(clampepsf(rw)) - sqrtf(clampepsf(t2));
    float eh = sqrtf(clampepsf(rh)) - sqrtf(clampepsf(t3));
    float loc = coo * (ex * ex + ey * ey + ew * ew + eh * eh);

    float not_contain = coo * nc * nc;

    // class loss: logsumexp(logits) - logits[cls]
    float m = L[0];
#pragma unroll
    for (int i = 1; i < 10; ++i) m = fmaxf(m, L[i]);
    float se = 0.0f;
#pragma unroll
    for (int i = 0; i < 10; ++i) se += expf(L[i] - m);
    float logz = logf(se) + m;
    int ci = (int)cls_t;
    float picked = L[0];
#pragma unroll
    for (int i = 1; i < 10; ++i) picked = (ci == i) ? L[i] : picked;
    float cls_loss = coo * (logz - picked);

    float cell_loss = 5.0f * loc + 2.0f * contain + not_contain +
                      0.5f * noobj + cls_loss;
    float acc = valid ? cell_loss : 0.0f;

    // wave32 reduction via FP32 WMMA, then block reduction through LDS
    float wsum = wave_sum32(acc);

    __shared__ float lds_part[8];
    int wid = threadIdx.x >> 5;
    int lane = threadIdx.x & 31;
    if (lane == 0) lds_part[wid] = wsum;
    __syncthreads();
    if (threadIdx.x == 0) {
        float s = 0.0f;
#pragma unroll
        for (int i = 0; i < 8; ++i) s += lds_part[i];
        blk_out[blockIdx.x] = s;
    }
}

// ---------------------------------------------------------------------------
// Pass 2: single block deterministically folds block partials, scales by 1/N.
// ---------------------------------------------------------------------------
__global__ void __launch_bounds__(1024)
yolo_reduce_kernel(const float* __restrict__ part, int n,
                   float* __restrict__ out, float invN) {
    float s = 0.0f;
    for (int i = threadIdx.x; i < n; i += 1024) s += part[i];

    float wsum = wave_sum32(s);   // converged point: EXEC all-1s

    __shared__ float lds_part[32];
    int wid = threadIdx.x >> 5;
    int lane = threadIdx.x & 31;
    if (lane == 0) lds_part[wid] = wsum;
    __syncthreads();
    if (threadIdx.x == 0) {
        float t = 0.0f;
#pragma unroll
        for (int i = 0; i < 32; ++i) t += lds_part[i];
        out[0] = t * invN;
    }
}

extern "C" void kernel_launch(void* const* d_in, const int* in_sizes, int n_in,
                              void* d_out, int out_size, void* d_ws, size_t ws_size,
                              hipStream_t stream) {
    const float* pred = (const float*)d_in[0];   // [N,7,7,20] f32
    const float* targ = (const float*)d_in[1];   // [N,7,7,6]  f32

    long long ncells = (long long)in_sizes[0] / 20;  // N*49
    long long N = ncells / 49;
    int nblocks = (int)((ncells + 255) / 256);
    float invN = 1.0f / (float)N;

    float* part = (float*)d_ws;  // nblocks floats of scratch

    yolo_cell_kernel<<<nblocks, 256, 0, stream>>>(pred, targ, part, ncells);
    yolo_reduce_kernel<<<1, 1024, 0, stream>>>(part, nblocks, (float*)d_out, invN);
}